// SEDR_49117245997357
// MI455X (gfx1250) — compile-verified
//
#include <hip/hip_runtime.h>
#include <hip/hip_bf16.h>

typedef __attribute__((ext_vector_type(2))) float v2f;
typedef __attribute__((ext_vector_type(8))) float v8f;

#define N_DRUG   50000
#define N_DIS    50000
#define NEDGE    800000
#define RNUM     2
#define IN_UNITS 2048
#define MSG      32
#define AGG      96       // 3 * MSG
#define OUTU     64
#define NBASIS   4
#define SLOPE    0.1f

// ---------------------------------------------------------------------------
// K1: W[r] = sum_b att[r,b] * basis[b]   -> (2, 2048, 32) f32
// ---------------------------------------------------------------------------
__global__ void gcmc_basis_kernel(const float* __restrict__ att,
                                  const float* __restrict__ basis,
                                  float* __restrict__ W) {
    const int per_r = IN_UNITS * MSG;                 // 65536
    int tid = blockIdx.x * blockDim.x + threadIdx.x;
    if (tid >= RNUM * per_r) return;
    int r  = tid / per_r;
    int im = tid - r * per_r;
    float acc = 0.f;
#pragma unroll
    for (int b = 0; b < NBASIS; ++b)
        acc += att[r * NBASIS + b] * basis[(size_t)b * per_r + im];
    W[tid] = acc;
}

// ---------------------------------------------------------------------------
// K2: fused gather + scatter-add.  One wave handles one edge at a time:
// lane l covers feature columns {l, l+32, l+64}.  The 3 gathers per edge are
// coalesced 128B rows of the L2-resident W[r] table.  grid.y selects
// (rating, direction).  Accumulates raw segment sums (ci applied later).
// ---------------------------------------------------------------------------
__global__ __launch_bounds__(256) void gcmc_scatter_kernel(
    const int*   __restrict__ edge_src,  // (R, E)
    const int*   __restrict__ edge_dst,  // (R, E)
    const int*   __restrict__ drug_idx,  // (N_DRUG, 3)
    const int*   __restrict__ dis_idx,   // (N_DIS, 3)
    const float* __restrict__ cj_drug,   // (N_DRUG,)
    const float* __restrict__ cj_dis,    // (N_DIS,)
    const float* __restrict__ W,         // (R, 2048, 32)
    float*       __restrict__ h_drug,    // (N_DRUG, 96)
    float*       __restrict__ h_dis) {   // (N_DIS, 96)
    const int lane = threadIdx.x & 31;
    const int wid  = threadIdx.x >> 5;

    const int combo = blockIdx.y;        // 0..3
    const int r   = combo >> 1;
    const int dir = combo & 1;

    const int* es = edge_src + (size_t)r * NEDGE;
    const int* ed = edge_dst + (size_t)r * NEDGE;

    const int*   src_nodes;  const int* dst_nodes;
    const int*   sidx;       const float* cj;  float* hdst;
    if (dir == 0) {          // drug(src) features -> h_dis[dst]
        src_nodes = es; dst_nodes = ed;
        sidx = drug_idx; cj = cj_drug; hdst = h_dis;
    } else {                 // dis(dst) features -> h_drug[src]
        src_nodes = ed; dst_nodes = es;
        sidx = dis_idx;  cj = cj_dis;  hdst = h_drug;
    }
    const float* Wr = W + (size_t)r * IN_UNITS * MSG;

#pragma unroll 1
    for (int t = 0; t < 4; ++t) {
        int e = (blockIdx.x * 8 + wid) * 4 + t;
        if (e >= NEDGE) continue;
        int s = src_nodes[e];
        int d = dst_nodes[e];
        float c = cj[s];
        float* hrow = hdst + (size_t)d * AGG;
#pragma unroll
        for (int j = 0; j < 3; ++j) {
            int row = sidx[s * 3 + j];
            float v = Wr[(size_t)row * MSG + lane] * c;
            (void)__hip_atomic_fetch_add(&hrow[j * MSG + lane], v,
                                         __ATOMIC_RELAXED,
                                         __HIP_MEMORY_SCOPE_AGENT);
        }
    }
}

// ---------------------------------------------------------------------------
// K3: out = LeakyReLU(h * ci) @ w_fc + b_fc  via V_WMMA_F32_16X16X4_F32.
// One wave -> one 16(M) x 64(N) tile; K=96 -> 24 steps of 4.
// A frag (16x4 f32): lanes 0-15 hold K={0,1}, lanes 16-31 hold K={2,3}.
// B frag (4x16 f32): VGPR0 = rows {K0,K2}, VGPR1 = rows {K1,K3}, N striped.
// D (16x16 f32): VGPR v: lanes 0-15 -> M=v, lanes 16-31 -> M=v+8.
// ---------------------------------------------------------------------------
#define HPAD 100   // 16x96 tile padded: stride 100 -> conflict-free A reads

__global__ __launch_bounds__(128) void gcmc_fc_wmma_kernel(
    const float* __restrict__ h_drug, const float* __restrict__ h_dis,
    const float* __restrict__ ci_drug, const float* __restrict__ ci_dis,
    const float* __restrict__ w_fc,   // (96, 64)
    const float* __restrict__ b_fc,   // (64,)
    float*       __restrict__ out) {  // (100000, 64): drug rows then dis rows
    __shared__ float lds_w[AGG * OUTU];      // 24 KB
    __shared__ float lds_h[4][16 * HPAD];    // 4 waves * 6.25 KB

    const int tid  = threadIdx.x;
    const int lane = tid & 31;
    const int wid  = tid >> 5;
    const int half = lane >> 4;
    const int m    = lane & 15;

    // stage w_fc cooperatively
    for (int i = tid; i < AGG * OUTU; i += 128) lds_w[i] = w_fc[i];

    const int n_tiles = (N_DRUG + N_DIS) / 16;     // 6250 (exact)
    const int tile = blockIdx.x * 4 + wid;         // wave-uniform
    const bool valid = (tile < n_tiles);

    const float* h = h_drug; const float* ci = ci_drug;
    float* o = out; int base = 0;
    if (valid) {
        if (tile < N_DRUG / 16) {
            base = tile * 16;
        } else {
            h = h_dis; ci = ci_dis;
            o = out + (size_t)N_DRUG * OUTU;
            base = (tile - N_DRUG / 16) * 16;
        }
        // stage 16x96 tile, applying ci scale + LeakyReLU(0.1), coalesced
#pragma unroll 1
        for (int i = 0; i < 48; ++i) {
            int idx  = i * 32 + lane;          // 0..1535
            int rrow = idx / AGG;
            int col  = idx - rrow * AGG;
            float v = h[(size_t)(base + rrow) * AGG + col] * ci[base + rrow];
            v = (v >= 0.f) ? v : SLOPE * v;
            lds_h[wid][rrow * HPAD + col] = v;
        }
    }
    __syncthreads();                                // uniform barrier
    if (!valid) return;                             // whole-wave exit: EXEC all-1s below

    v8f acc0 = {}, acc1 = {}, acc2 = {}, acc3 = {};
#pragma unroll 1
    for (int kk = 0; kk < AGG / 4; ++kk) {          // 24 k-steps
        const int kb = kk * 4;
        v2f a;
        a.x = lds_h[wid][m * HPAD + kb + 2 * half];
        a.y = lds_h[wid][m * HPAD + kb + 2 * half + 1];
        v2f b0, b1, b2, b3;
        b0.x = lds_w[(kb + 2 * half) * OUTU +  0 + m];
        b0.y = lds_w[(kb + 2 * half + 1) * OUTU +  0 + m];
        b1.x = lds_w[(kb + 2 * half) * OUTU + 16 + m];
        b1.y = lds_w[(kb + 2 * half + 1) * OUTU + 16 + m];
        b2.x = lds_w[(kb + 2 * half) * OUTU + 32 + m];
        b2.y = lds_w[(kb + 2 * half + 1) * OUTU + 32 + m];
        b3.x = lds_w[(kb + 2 * half) * OUTU + 48 + m];
        b3.y = lds_w[(kb + 2 * half + 1) * OUTU + 48 + m];
        acc0 = __builtin_amdgcn_wmma_f32_16x16x4_f32(false, a, false, b0,
                                                     (short)0, acc0, false, false);
        acc1 = __builtin_amdgcn_wmma_f32_16x16x4_f32(false, a, false, b1,
                                                     (short)0, acc1, false, false);
        acc2 = __builtin_amdgcn_wmma_f32_16x16x4_f32(false, a, false, b2,
                                                     (short)0, acc2, false, false);
        acc3 = __builtin_amdgcn_wmma_f32_16x16x4_f32(false, a, false, b3,
                                                     (short)0, acc3, false, false);
    }

    // writeback: D VGPR v -> row (base + v + 8*half), col (nt*16 + m)
    v8f accs[4] = {acc0, acc1, acc2, acc3};
#pragma unroll
    for (int nt = 0; nt < 4; ++nt) {
        float bias = b_fc[nt * 16 + m];
#pragma unroll
        for (int v = 0; v < 8; ++v) {
            int row = base + v + 8 * half;
            o[(size_t)row * OUTU + nt * 16 + m] = accs[nt][v] + bias;
        }
    }
}

// ---------------------------------------------------------------------------
// launcher
// ---------------------------------------------------------------------------
extern "C" void kernel_launch(void* const* d_in, const int* in_sizes, int n_in,
                              void* d_out, int out_size, void* d_ws, size_t ws_size,
                              hipStream_t stream) {
    const int*   drug_idx = (const int*)  d_in[0];
    const int*   dis_idx  = (const int*)  d_in[1];
    const int*   edge_src = (const int*)  d_in[2];
    const int*   edge_dst = (const int*)  d_in[3];
    const float* cj_drug  = (const float*)d_in[4];
    const float* ci_drug  = (const float*)d_in[5];
    const float* cj_dis   = (const float*)d_in[6];
    const float* ci_dis   = (const float*)d_in[7];
    const float* att      = (const float*)d_in[8];
    const float* basis    = (const float*)d_in[9];
    const float* w_fc     = (const float*)d_in[10];
    const float* b_fc     = (const float*)d_in[11];
    float* out = (float*)d_out;

    // workspace layout (all 256B aligned):
    //   W      : 2*2048*32 f32      = 524,288 B
    //   h_drug : 50000*96  f32      = 19,200,000 B
    //   h_dis  : 50000*96  f32      = 19,200,000 B
    char* ws = (char*)d_ws;
    float* W      = (float*)(ws);
    float* h_drug = (float*)(ws + (size_t)RNUM * IN_UNITS * MSG * 4);
    float* h_dis  = (float*)(ws + (size_t)RNUM * IN_UNITS * MSG * 4
                                + (size_t)N_DRUG * AGG * 4);

    // zero accumulators (graph-capture-safe async memset)
    hipMemsetAsync(h_drug, 0, (size_t)(N_DRUG + N_DIS) * AGG * sizeof(float), stream);

    // K1: basis-weighted transforms
    {
        int total = RNUM * IN_UNITS * MSG;   // 131072
        gcmc_basis_kernel<<<(total + 255) / 256, 256, 0, stream>>>(att, basis, W);
    }

    // K2: edge scatter (grid.y = rating x direction; 32 edges per block)
    {
        dim3 grid((NEDGE + 31) / 32, RNUM * 2);
        gcmc_scatter_kernel<<<grid, 256, 0, stream>>>(
            edge_src, edge_dst, drug_idx, dis_idx,
            cj_drug, cj_dis, W, h_drug, h_dis);
    }

    // K3: activation + FC via f32 WMMA (one wave per 16x64 tile)
    {
        int n_tiles = (N_DRUG + N_DIS) / 16;   // 6250
        gcmc_fc_wmma_kernel<<<(n_tiles + 3) / 4, 128, 0, stream>>>(
            h_drug, h_dis, ci_drug, ci_dis, w_fc, b_fc, out);
    }
}